// GCNPool_my3_44727789421338
// MI455X (gfx1250) — compile-verified
//
#include <hip/hip_runtime.h>
#include <hip/hip_bf16.h>
#include <math.h>

// ---------------------------------------------------------------------------
// GCNPool forward for MI455X (gfx1250, wave32).
// Dominant cost: six 4000x4000x3072 fp32 GEMMs -> V_WMMA_F32_16X16X4_F32,
// fed by double-buffered GLOBAL_LOAD_ASYNC_TO_LDS_B128 staging (ASYNCcnt).
// A (64MB) + X (49MB) fit in 192MB L2 -> compute-bound; fp32 WMMA keeps
// reference precision (RNE) while using the matrix pipe.
// ---------------------------------------------------------------------------

#define Bb 8
#define Cc 32
#define Nn 4000
#define Tt 12
#define TSs 6
#define Mm (Bb * Cc * Tt)       // 3072 columns of the v-major matrices

typedef float v2f __attribute__((ext_vector_type(2)));
typedef float v8f __attribute__((ext_vector_type(8)));
typedef int   v4i_vs __attribute__((vector_size(16)));   // matches builtin param

// ---------------- CDNA5 async global->LDS helpers ---------------------------
#if defined(__has_builtin)
#  if __has_builtin(__builtin_amdgcn_global_load_async_to_lds_b128)
#    define HAVE_ASYNC_LDS 1
#  endif
#  if __has_builtin(__builtin_amdgcn_s_wait_asynccnt)
#    define HAVE_WAIT_ASYNC 1
#  endif
#endif

__device__ __forceinline__ void copy16_async(const float* g, float* l) {
#ifdef HAVE_ASYNC_LDS
    // per-lane 16B DMA: global -> LDS, tracked by ASYNCcnt
    __builtin_amdgcn_global_load_async_to_lds_b128(
        (__attribute__((address_space(1))) v4i_vs*)g,
        (__attribute__((address_space(3))) v4i_vs*)l,
        0 /*offset*/, 0 /*cpol*/);
#else
    *(float4*)l = *(const float4*)g;
#endif
}

__device__ __forceinline__ void wait_async_lds() {
#ifdef HAVE_ASYNC_LDS
#  ifdef HAVE_WAIT_ASYNC
    __builtin_amdgcn_s_wait_asynccnt(0);
#  else
    asm volatile("s_wait_asynccnt 0" ::: "memory");
#  endif
#endif
}

// ---------------- workspace layout (float offsets) -------------------------
#define GRP_ELEMS ((size_t)Nn * Mm)            // 12,288,000
#define OFF_XT    ((size_t)0)
#define OFF_G(s)  (GRP_ELEMS * (size_t)(1 + (s)))   // s = 0..5
#define OFF_WT    (GRP_ELEMS * 7)                    // 86,016,000
#define WT_ELEMS  (672 * 64)
#define OFF_RES6  (OFF_WT + WT_ELEMS)
#define R6_ELEMS  ((size_t)Bb * Nn * Cc * TSs)       // 6,144,000  (B,N,C,6)
#define OFF_G2    (OFF_RES6 + R6_ELEMS)
#define OFF_YB    (OFF_G2 + R6_ELEMS)
#define OFF_F1    (OFF_YB + R6_ELEMS)
#define F1_ELEMS  ((size_t)Bb * TSs * Nn)            // 192,000
#define OFF_U     (OFF_F1 + F1_ELEMS)
#define OFF_F2    (OFF_U + 1536)
#define OFF_COEF  (OFF_F2 + 1536)
#define OFF_STATS (OFF_COEF + 288)
#define OFF_SCALE (OFF_STATS + 64)

// ---------------------------------------------------------------------------
__global__ __launch_bounds__(64) void zero_stats_kernel(float* stats) {
    if (threadIdx.x < 64) stats[threadIdx.x] = 0.0f;
}

// x (B,C,N,T) -> Xt[v][m], m = (b*C+c)*T + l  (v-major for the GEMMs)
__global__ __launch_bounds__(256) void transpose_x_kernel(const float* __restrict__ x,
                                                          float* __restrict__ Xt) {
    size_t i = (size_t)blockIdx.x * 256 + threadIdx.x;
    if (i >= GRP_ELEMS) return;
    int v = (int)(i / Mm);
    int m = (int)(i % Mm);
    Xt[i] = x[(size_t)(m / Tt) * ((size_t)Nn * Tt) + (size_t)v * Tt + (m % Tt)];
}

// residual6[b,n,o,q] = conv1_b[o] + sum_c conv1_w[o,c] * x[b,c,n,6+q]
__global__ __launch_bounds__(192) void residual_kernel(const float* __restrict__ x,
                                                       const float* __restrict__ w,
                                                       const float* __restrict__ bias,
                                                       float* __restrict__ res6) {
    int bn = blockIdx.x;
    int b = bn / Nn, n = bn % Nn;
    __shared__ float xs[Cc * TSs];
    int t = threadIdx.x;                           // 0..191
    int o = t / TSs, q = t % TSs;
    xs[t] = x[((size_t)(b * Cc + o) * Nn + n) * Tt + (Tt - TSs) + q];
    __syncthreads();
    float acc = bias[o];
    for (int ci = 0; ci < Cc; ++ci) acc += w[o * Cc + ci] * xs[ci * TSs + q];
    res6[(size_t)bn * (Cc * TSs) + t] = acc;
}

// ---------------------------------------------------------------------------
// Y[r][m] = sum_v A[r][v] * X[v][m]   (A: 4000x4000, X: 4000x3072, row-major)
// 80x128 tile per 256-thread workgroup, K-chunks of 32, f32 WMMA 16x16x4.
// Double-buffered LDS; tiles streamed via async global->LDS B128 copies.
// ---------------------------------------------------------------------------
#define APITCH 36     // 36*4 = 144B per row: 16B aligned rows, ds_b64-aligned frags
#define XPITCH 132    // 132*4 = 528B per row: 16B aligned rows

__global__ __launch_bounds__(256) void gemm_aX_wmma_kernel(const float* __restrict__ A,
                                                           const float* __restrict__ X,
                                                           float* __restrict__ Y) {
    __shared__ float As[2][80 * APITCH];   // [row][k]
    __shared__ float Xs[2][32 * XPITCH];   // [k][m]
    const int tid  = threadIdx.x;
    const int wave = tid >> 5;
    const int lane = tid & 31;
    const int rowBase = blockIdx.y * 80;
    const int colBase = blockIdx.x * 128;
    const int laneM = lane & 15;
    const int kOff  = (lane >> 4) * 2;   // lanes 0-15 -> K 0..1, lanes 16-31 -> K 2..3

    v8f acc[5];
    #pragma unroll
    for (int r = 0; r < 5; ++r) acc[r] = (v8f){0, 0, 0, 0, 0, 0, 0, 0};

    // issue one K-chunk's tiles (A: 80x32, X: 32x128) as 16B-per-lane copies
#define STAGE(K0, BUF)                                                          \
    do {                                                                        \
        for (int i_ = tid; i_ < 640; i_ += 256) {                               \
            int r_ = i_ >> 3, k4_ = (i_ & 7) << 2;                              \
            copy16_async(&A[(size_t)(rowBase + r_) * Nn + (K0) + k4_],          \
                         &As[BUF][r_ * APITCH + k4_]);                          \
        }                                                                       \
        for (int i_ = tid; i_ < 1024; i_ += 256) {                              \
            int kk_ = i_ >> 5, m4_ = (i_ & 31) << 2;                            \
            copy16_async(&X[(size_t)((K0) + kk_) * Mm + colBase + m4_],         \
                         &Xs[BUF][kk_ * XPITCH + m4_]);                         \
        }                                                                       \
    } while (0)

    STAGE(0, 0);
    wait_async_lds();
    __syncthreads();

    const int colW = wave * 16 + laneM;
    const int nChunks = Nn / 32;          // 125
    for (int chunk = 0; chunk < nChunks; ++chunk) {
        const int cur = chunk & 1;
        if (chunk + 1 < nChunks) STAGE((chunk + 1) * 32, cur ^ 1);  // overlap DMA

        const float* aBase = &As[cur][laneM * APITCH + kOff];
        const float* xBase = &Xs[cur][kOff * XPITCH + colW];
        #pragma unroll
        for (int ks = 0; ks < 32; ks += 4) {
            v2f bf;
            bf.x = xBase[ks * XPITCH];          // lane-consecutive cols: no conflicts
            bf.y = xBase[(ks + 1) * XPITCH];
            #pragma unroll
            for (int rt = 0; rt < 5; ++rt) {
                v2f af = *(const v2f*)(aBase + rt * 16 * APITCH + ks);
                acc[rt] = __builtin_amdgcn_wmma_f32_16x16x4_f32(
                    /*neg_a=*/false, af, /*neg_b=*/false, bf,
                    /*c_mod=*/(short)0, acc[rt],
                    /*reuse_a=*/false, /*reuse_b=*/false);
            }
        }
        wait_async_lds();    // own async copies done ...
        __syncthreads();     // ... all waves' copies done, cur free for reuse
    }
#undef STAGE

    // C/D layout: VGPR i -> M=i (lanes 0-15) / M=i+8 (lanes 16-31); N = laneM
    const int col   = colBase + colW;
    const int mHalf = (lane >> 4) * 8;
    #pragma unroll
    for (int rt = 0; rt < 5; ++rt) {
        int r0 = rowBase + rt * 16 + mHalf;
        #pragma unroll
        for (int i = 0; i < 8; ++i)
            Y[(size_t)(r0 + i) * Mm + col] = acc[rt][i];
    }
}

// mlp_w (64,224,1,3) -> Wt[k=ic*3+kt][o]  (coalesced weight reads later)
__global__ __launch_bounds__(256) void wt_kernel(const float* __restrict__ mlp_w,
                                                 float* __restrict__ Wt) {
    int i = blockIdx.x * 256 + threadIdx.x;
    if (i >= WT_ELEMS) return;
    int o = i % 64, k = i / 64;
    Wt[i] = mlp_w[o * 672 + k];
}

// Fused: mlp conv (224->64, kernel 1x3) + tanh*sigmoid gate + time-conv (10->6)
// One block per (b,n); 320 threads = 10 waves; per wave: j uniform, o = lane.
__global__ __launch_bounds__(320) void mlp_gate_ct1_kernel(
        const float* __restrict__ g0, const float* __restrict__ g1,
        const float* __restrict__ g2a, const float* __restrict__ g3,
        const float* __restrict__ g4, const float* __restrict__ g5,
        const float* __restrict__ g6,
        const float* __restrict__ Wt, const float* __restrict__ mlp_b,
        const float* __restrict__ ct1_w, const float* __restrict__ ct1_b,
        float* __restrict__ g2out) {
    __shared__ float in7[7 * 384];    // concat channel block for this (b,n)
    __shared__ float gbuf[320];       // gated activations g[c][j]
    __shared__ float sctw[60];
    const int bn = blockIdx.x;
    const int b = bn / Nn, n = bn % Nn;
    const int t = threadIdx.x;
    const float* gp[7] = {g0, g1, g2a, g3, g4, g5, g6};
    const size_t colbase = (size_t)n * Mm + (size_t)b * (Cc * Tt);
    for (int g = 0; g < 7; ++g)
        for (int idx = t; idx < 384; idx += 320)
            in7[g * 384 + idx] = gp[g][colbase + idx];   // contiguous 384 floats
    if (t < 60) sctw[t] = ct1_w[t];
    __syncthreads();

    const int o = t & 31;     // lane
    const int j = t >> 5;     // wave-uniform, 0..9
    float a1 = mlp_b[o], a2 = mlp_b[o + 32];
    for (int gc = 0; gc < 224; ++gc) {
        const float* xr = &in7[gc * 12 + j];
        const float* wr = &Wt[gc * 3 * 64 + o];
        #pragma unroll
        for (int kt = 0; kt < 3; ++kt) {
            float xv = xr[kt];                    // LDS broadcast (wave-uniform)
            a1 = fmaf(wr[kt * 64], xv, a1);       // coalesced 128B
            a2 = fmaf(wr[kt * 64 + 32], xv, a2);  // coalesced 128B
        }
    }
    float gv = tanhf(a1) * (1.0f / (1.0f + expf(-a2)));
    gbuf[o * 10 + j] = gv;
    __syncthreads();

    if (t < Cc * TSs) {                            // 192 outputs, coalesced store
        int c = t / TSs, q = t % TSs;
        float acc = ct1_b[q];
        #pragma unroll
        for (int jj = 0; jj < 10; ++jj) acc += sctw[q * 10 + jj] * gbuf[c * 10 + jj];
        g2out[(size_t)bn * (Cc * TSs) + t] = acc;  // layout (B,N,C,6)
    }
}

// f1[b,t,n] = sum_c tat_c1_w[c] * g2[b,n,c,t]
__global__ __launch_bounds__(256) void f1_kernel(const float* __restrict__ g2,
                                                 const float* __restrict__ c1w,
                                                 float* __restrict__ f1) {
    int i = blockIdx.x * 256 + threadIdx.x;
    if (i >= (int)F1_ELEMS) return;
    int n = i % Nn; int r = i / Nn; int tt = r % TSs; int b = r / TSs;
    size_t base = ((size_t)(b * Nn + n)) * (Cc * TSs) + tt;
    float acc = 0.f;
    for (int c = 0; c < Cc; ++c) acc += c1w[c] * g2[base + c * TSs];
    f1[i] = acc;                                    // layout (B,6,N)
}

// u[b,t,c] = sum_n f1[b,t,n] * tat_w[n,c]   (block reduction over n)
__global__ __launch_bounds__(256) void u_kernel(const float* __restrict__ f1,
                                                const float* __restrict__ tat_w,
                                                float* __restrict__ u) {
    int blk = blockIdx.x;                 // 8*6*32 = 1536
    int c = blk & 31; int r = blk >> 5; int tt = r % TSs; int b = r / TSs;
    int t = threadIdx.x;
    const float* f1r = f1 + (size_t)(b * TSs + tt) * Nn;
    float acc = 0.f;
    for (int n = t; n < Nn; n += 256) acc += f1r[n] * tat_w[n * Cc + c];
    __shared__ float red[256];
    red[t] = acc; __syncthreads();
    for (int s = 128; s > 0; s >>= 1) { if (t < s) red[t] += red[t + s]; __syncthreads(); }
    if (t == 0) u[(b * TSs + tt) * Cc + c] = red[0];
}

// f2[b,c,t] = sum_n tat_c2_w[n] * g2[b,n,c,t]
__global__ __launch_bounds__(256) void f2_kernel(const float* __restrict__ g2,
                                                 const float* __restrict__ c2w,
                                                 float* __restrict__ f2) {
    int blk = blockIdx.x;                 // 8*32*6 = 1536
    int b = blk / (Cc * TSs); int rem = blk % (Cc * TSs);
    int c = rem / TSs; int tt = rem % TSs;
    int t = threadIdx.x;
    float acc = 0.f;
    for (int n = t; n < Nn; n += 256)
        acc += c2w[n] * g2[((size_t)(b * Nn + n) * Cc + c) * TSs + tt];
    __shared__ float red[256];
    red[t] = acc; __syncthreads();
    for (int s = 128; s > 0; s >>= 1) { if (t < s) red[t] += red[t + s]; __syncthreads(); }
    if (t == 0) f2[(b * Cc + c) * TSs + tt] = red[0];
}

// Single block: logits -> sigmoid -> tat_v mix -> BN1d(tem_size) -> softmax
__global__ __launch_bounds__(384) void attn_kernel(const float* __restrict__ u,
                                                   const float* __restrict__ f2,
                                                   const float* __restrict__ tat_bias,
                                                   const float* __restrict__ tat_v,
                                                   const float* __restrict__ bn1_g,
                                                   const float* __restrict__ bn1_b,
                                                   float* __restrict__ coefs) {
    __shared__ float su[1536], sf2[1536], s1[288], lg[288], ln[288], mu[6], rs[6];
    int t = threadIdx.x;
    for (int i = t; i < 1536; i += 384) { su[i] = u[i]; sf2[i] = f2[i]; }
    __syncthreads();
    if (t < 288) {                                   // logits[b,t,s] then sigmoid
        int b = t / 36, tt = (t / 6) % 6, s = t % 6;
        float acc = tat_bias[tt * 6 + s];
        for (int c = 0; c < Cc; ++c)
            acc += su[(b * TSs + tt) * Cc + c] * sf2[(b * Cc + c) * TSs + s];
        s1[t] = 1.0f / (1.0f + expf(-acc));
    }
    __syncthreads();
    if (t < 288) {                                   // btq = sum_s tat_v[t,s]*sig[b,s,q]
        int b = t / 36, tt = (t / 6) % 6, q = t % 6;
        float acc = 0.f;
        for (int s = 0; s < 6; ++s) acc += tat_v[tt * 6 + s] * s1[b * 36 + s * 6 + q];
        lg[t] = acc;
    }
    __syncthreads();
    if (t < 6) {                                     // BN over (b,t) per channel q
        float s = 0.f, s2 = 0.f;
        for (int b = 0; b < Bb; ++b)
            for (int tt = 0; tt < 6; ++tt) {
                float v = lg[b * 36 + tt * 6 + t];
                s += v; s2 += v * v;
            }
        float m = s / 48.0f;
        float var = s2 / 48.0f - m * m;
        mu[t] = m; rs[t] = rsqrtf(var + 1e-5f);
    }
    __syncthreads();
    if (t < 288) {
        int q = t % 6;
        ln[t] = (lg[t] - mu[q]) * rs[q] * bn1_g[q] + bn1_b[q];
    }
    __syncthreads();
    if (t < 48) {                                    // softmax over q per (b,t)
        int base = t * 6;
        float mx = ln[base];
        for (int q = 1; q < 6; ++q) mx = fmaxf(mx, ln[base + q]);
        float e[6], sum = 0.f;
        for (int q = 0; q < 6; ++q) { e[q] = expf(ln[base + q] - mx); sum += e[q]; }
        for (int q = 0; q < 6; ++q) coefs[base + q] = e[q] / sum;   // [b][t][q]
    }
}

// y = out + residual, plus BN2 batch-stats partials (LDS-reduced atomics)
__global__ __launch_bounds__(192) void combine_kernel(const float* __restrict__ g2,
                                                      const float* __restrict__ res6,
                                                      const float* __restrict__ coefs,
                                                      float* __restrict__ ybuf,
                                                      float* __restrict__ stats) {
    int b = blockIdx.x >> 5;
    int chunk = blockIdx.x & 31;               // 32 chunks x 125 n
    int t = threadIdx.x;                       // 0..191 ; c = t/6, q = t%6
    __shared__ float sc[36];
    __shared__ float sred[192], sred2[192];
    if (t < 36) sc[t] = coefs[b * 36 + t];
    __syncthreads();
    int c = t / TSs, q = t % TSs;
    float s = 0.f, s2 = 0.f;
    for (int nn = 0; nn < 125; ++nn) {
        int n = chunk * 125 + nn;
        size_t base = (size_t)(b * Nn + n) * (Cc * TSs);
        const float* grow = g2 + base + c * TSs;
        float y = res6[base + t];
        #pragma unroll
        for (int l = 0; l < 6; ++l) y += grow[l] * sc[q * 6 + l];  // coefs[b,q,l]
        ybuf[base + t] = y;
        s += y; s2 += y * y;
    }
    sred[t] = s; sred2[t] = s2;
    __syncthreads();
    if (q == 0) {
        float a = 0.f, a2 = 0.f;
        #pragma unroll
        for (int l = 0; l < 6; ++l) { a += sred[c * 6 + l]; a2 += sred2[c * 6 + l]; }
        atomicAdd(&stats[c], a);
        atomicAdd(&stats[32 + c], a2);
    }
}

__global__ __launch_bounds__(64) void finalize_stats_kernel(const float* __restrict__ stats,
                                                            const float* __restrict__ bn2_g,
                                                            const float* __restrict__ bn2_b,
                                                            float* __restrict__ scale) {
    int c = threadIdx.x;
    if (c < Cc) {
        const float cnt = (float)(Bb * Nn * TSs);  // 192000
        float m = stats[c] / cnt;
        float var = stats[32 + c] / cnt - m * m;
        float a = bn2_g[c] * rsqrtf(var + 1e-5f);
        scale[c] = a;
        scale[32 + c] = bn2_b[c] - m * a;
    }
}

// (B,N,C,6) ybuf -> (B,C,N,6) normalized output
__global__ __launch_bounds__(256) void output_kernel(const float* __restrict__ ybuf,
                                                     const float* __restrict__ scale,
                                                     float* __restrict__ out) {
    size_t i = (size_t)blockIdx.x * 256 + threadIdx.x;
    if (i >= R6_ELEMS) return;
    int q = (int)(i % TSs);
    size_t r = i / TSs;
    int n = (int)(r % Nn);
    size_t r2 = r / Nn;
    int c = (int)(r2 % Cc);
    int b = (int)(r2 / Cc);
    size_t yi = ((size_t)(b * Nn + n) * Cc + c) * TSs + q;
    out[i] = scale[c] * ybuf[yi] + scale[32 + c];
}

// ---------------------------------------------------------------------------
extern "C" void kernel_launch(void* const* d_in, const int* in_sizes, int n_in,
                              void* d_out, int out_size, void* d_ws, size_t ws_size,
                              hipStream_t stream) {
    const float* x       = (const float*)d_in[0];
    const float* A0      = (const float*)d_in[1];
    const float* A1      = (const float*)d_in[2];
    const float* A2      = (const float*)d_in[3];
    const float* conv1_w = (const float*)d_in[4];
    const float* conv1_b = (const float*)d_in[5];
    const float* mlp_w   = (const float*)d_in[6];
    const float* mlp_b   = (const float*)d_in[7];
    const float* ct1_w   = (const float*)d_in[8];
    const float* ct1_b   = (const float*)d_in[9];
    const float* c1w     = (const float*)d_in[10];
    const float* c2w     = (const float*)d_in[11];
    const float* tat_w   = (const float*)d_in[12];
    const float* tat_bias= (const float*)d_in[13];
    const float* tat_v   = (const float*)d_in[14];
    const float* bn1_g   = (const float*)d_in[15];
    const float* bn1_b   = (const float*)d_in[16];
    const float* bn2_g   = (const float*)d_in[17];
    const float* bn2_b   = (const float*)d_in[18];
    float* out = (float*)d_out;
    float* ws  = (float*)d_ws;

    float* Xt    = ws + OFF_XT;
    float* G[6];
    for (int s = 0; s < 6; ++s) G[s] = ws + OFF_G(s);
    float* Wt    = ws + OFF_WT;
    float* res6  = ws + OFF_RES6;
    float* g2    = ws + OFF_G2;
    float* ybuf  = ws + OFF_YB;
    float* f1    = ws + OFF_F1;
    float* u     = ws + OFF_U;
    float* f2    = ws + OFF_F2;
    float* coefs = ws + OFF_COEF;
    float* stats = ws + OFF_STATS;
    float* scale = ws + OFF_SCALE;

    zero_stats_kernel<<<1, 64, 0, stream>>>(stats);

    transpose_x_kernel<<<(int)((GRP_ELEMS + 255) / 256), 256, 0, stream>>>(x, Xt);

    residual_kernel<<<Bb * Nn, 192, 0, stream>>>(x, conv1_w, conv1_b, res6);

    // six 4000x4000x3072 fp32 WMMA GEMMs (L2-resident operands, async staging)
    dim3 ggrid(Mm / 128, Nn / 80);
    gemm_aX_wmma_kernel<<<ggrid, 256, 0, stream>>>(A0, Xt,  G[0]);
    gemm_aX_wmma_kernel<<<ggrid, 256, 0, stream>>>(A0, G[0], G[1]);
    gemm_aX_wmma_kernel<<<ggrid, 256, 0, stream>>>(A1, Xt,  G[2]);
    gemm_aX_wmma_kernel<<<ggrid, 256, 0, stream>>>(A1, G[2], G[3]);
    gemm_aX_wmma_kernel<<<ggrid, 256, 0, stream>>>(A2, Xt,  G[4]);
    gemm_aX_wmma_kernel<<<ggrid, 256, 0, stream>>>(A2, G[4], G[5]);

    wt_kernel<<<(WT_ELEMS + 255) / 256, 256, 0, stream>>>(mlp_w, Wt);

    mlp_gate_ct1_kernel<<<Bb * Nn, 320, 0, stream>>>(
        Xt, G[0], G[1], G[2], G[3], G[4], G[5],
        Wt, mlp_b, ct1_w, ct1_b, g2);

    f1_kernel<<<(int)((F1_ELEMS + 255) / 256), 256, 0, stream>>>(g2, c1w, f1);
    u_kernel<<<Bb * TSs * Cc, 256, 0, stream>>>(f1, tat_w, u);
    f2_kernel<<<Bb * Cc * TSs, 256, 0, stream>>>(g2, c2w, f2);
    attn_kernel<<<1, 384, 0, stream>>>(u, f2, tat_bias, tat_v, bn1_g, bn1_b, coefs);

    combine_kernel<<<Bb * 32, 192, 0, stream>>>(g2, res6, coefs, ybuf, stats);
    finalize_stats_kernel<<<1, 64, 0, stream>>>(stats, bn2_g, bn2_b, scale);
    output_kernel<<<(int)((R6_ELEMS + 255) / 256), 256, 0, stream>>>(ybuf, scale, out);
}